// YOLOLoss_69535520522282
// MI455X (gfx1250) — compile-verified
//
#include <hip/hip_runtime.h>
#include <math.h>

// Problem constants (from reference)
#define Bc    16
#define Ac    3
#define Cc    80
#define Hc    80
#define Wc    80
#define Nc    512
#define STRIDEc 8.0f          // IMG / H = 640 / 80
#define CHc   (5 + Cc)        // 85
#define PLANEc (Hc * Wc)      // 6400
#define NCELL (Bc * Ac * Hc * Wc)   // 307200
#define NWORDS (NCELL / 32)         // 9600 bitmask words
#define NVEC  (NCELL / 4)           // 76800 float4 chunks

// Workspace float layout:
//  [0 .. 299]   : obj block partials (300 blocks)
//  [300 .. 301] : cls block partials (2 blocks)
//  [302 .. 303] : box block partials (2 blocks)
//  byte offset 1216: bit-packed target_obj mask, NWORDS uint32 (38400 B)
#define OBJ_BLOCKS 300
#define WS_MASK_BYTE_OFF 1216

typedef __attribute__((ext_vector_type(2))) float v2f;
typedef __attribute__((ext_vector_type(4))) float v4f;
typedef __attribute__((ext_vector_type(8))) float v8f;

// Wave32 sum reduction via V_WMMA_F32_16X16X4_F32.
// A holds one partial per lane (second slot zero); B is all-ones, so
// D[m][n] = rowsum(A[m]).  Per-lane sum of the 8 C/D VGPRs gives the sum over
// half the rows (lanes 0-15: m=0..7, lanes 16-31: m=8..15); one shfl_xor(16)
// completes the full 32-lane sum, replicated to every lane.
// Requires EXEC == all ones: call sites have uniform control flow.
__device__ __forceinline__ float wave_sum_wmma(float v) {
  v2f a; a[0] = v;    a[1] = 0.0f;
  v2f b; b[0] = 1.0f; b[1] = 1.0f;
  v8f c = {};
  c = __builtin_amdgcn_wmma_f32_16x16x4_f32(false, a, false, b,
                                            (short)0, c, false, false);
  float s = c[0] + c[1] + c[2] + c[3] + c[4] + c[5] + c[6] + c[7];
  s += __shfl_xor(s, 16, 32);
  return s;
}

__device__ __forceinline__ float bce_logits(float x, float y) {
  return fmaxf(x, 0.0f) - x * y + log1pf(expf(-fabsf(x)));
}

__global__ void zero_ws_kernel(unsigned int* __restrict__ mask) {
  int tid = blockIdx.x * blockDim.x + threadIdx.x;
  int stride = gridDim.x * blockDim.x;
  for (int i = tid; i < NWORDS; i += stride) mask[i] = 0u;
}

// One thread per target (512 = 2 blocks x 256, no partial blocks).
__global__ void per_target_kernel(const float* __restrict__ pred,
                                  const float* __restrict__ targets,
                                  const float* __restrict__ anchors,
                                  unsigned int* __restrict__ mask,
                                  float* __restrict__ partials) {
  __shared__ float smem[16];
  int n = blockIdx.x * blockDim.x + threadIdx.x;

  const float* t = targets + n * 6;
  int   b_idx = (int)t[0];
  int   t_cls = (int)t[1];
  float x1 = t[2], y1 = t[3], x2 = t[4], y2 = t[5];
  float tcx = 0.5f * (x1 + x2), tcy = 0.5f * (y1 + y2);
  float tw  = x2 - x1,          th  = y2 - y1;

  int gi = (int)(tcx / STRIDEc); gi = gi < 0 ? 0 : (gi > Wc - 1 ? Wc - 1 : gi);
  int gj = (int)(tcy / STRIDEc); gj = gj < 0 ? 0 : (gj > Hc - 1 ? Hc - 1 : gj);

  // Best anchor: argmin over max-ratio score; first-min tie-break (jnp.argmin).
  // (t_wh/stride)/(anchors/stride) == t_wh/anchors.
  float best_score = 3.4e38f; int best = 0;
  for (int a2 = 0; a2 < Ac; ++a2) {
    float aw = anchors[2 * a2 + 0], ah = anchors[2 * a2 + 1];
    float rw = tw / aw; rw = fmaxf(rw, 1.0f / rw);
    float rh = th / ah; rh = fmaxf(rh, 1.0f / rh);
    float sc = fmaxf(rw, rh);
    bool better = sc < best_score;
    best       = better ? a2 : best;
    best_score = better ? sc : best_score;
  }

  int cell = ((b_idx * Ac + best) * Hc + gj) * Wc + gi;
  atomicOr(&mask[cell >> 5], 1u << (cell & 31));   // idempotent like .set(1.0)

  long base = ((long)(b_idx * Ac + best) * CHc) * PLANEc + (long)(gj * Wc + gi);

  // Classification BCE vs one-hot over 80 classes (uniform trip count).
  float cls_sum = 0.0f;
  for (int c2 = 0; c2 < Cc; ++c2) {
    float x = pred[base + (long)(5 + c2) * PLANEc];
    float y = (c2 == t_cls) ? 1.0f : 0.0f;
    cls_sum += bce_logits(x, y);
  }

  // Box decode + CIoU.
  float tx  = pred[base + 0L * PLANEc];
  float ty  = pred[base + 1L * PLANEc];
  float twp = pred[base + 2L * PLANEc];
  float thp = pred[base + 3L * PLANEc];
  float px = (1.0f / (1.0f + expf(-tx)) + (float)gi) * STRIDEc;
  float py = (1.0f / (1.0f + expf(-ty)) + (float)gj) * STRIDEc;
  float pw = expf(twp) * anchors[2 * best + 0] * STRIDEc;
  float ph = expf(thp) * anchors[2 * best + 1] * STRIDEc;

  float b1x1 = px - 0.5f * pw, b1y1 = py - 0.5f * ph;
  float b1x2 = px + 0.5f * pw, b1y2 = py + 0.5f * ph;
  float b2x1 = tcx - 0.5f * tw, b2y1 = tcy - 0.5f * th;
  float b2x2 = tcx + 0.5f * tw, b2y2 = tcy + 0.5f * th;
  float iw = fmaxf(fminf(b1x2, b2x2) - fmaxf(b1x1, b2x1), 0.0f);
  float ih = fmaxf(fminf(b1y2, b2y2) - fmaxf(b1y1, b2y1), 0.0f);
  float inter = iw * ih;
  float area1 = (b1x2 - b1x1) * (b1y2 - b1y1);
  float area2 = (b2x2 - b2x1) * (b2y2 - b2y1);
  float uni   = area1 + area2 - inter + 1e-10f;
  float iou   = inter / uni;
  float cd    = (px - tcx) * (px - tcx) + (py - tcy) * (py - tcy);
  float ew = fmaxf(b1x2, b2x2) - fminf(b1x1, b2x1);
  float eh = fmaxf(b1y2, b2y2) - fminf(b1y1, b2y1);
  float diag = ew * ew + eh * eh + 1e-10f;
  float da = atanf(pw / ph) - atanf(tw / th);
  float v  = 0.405284734569351086f * da * da;     // 4/pi^2
  float alpha = v / (1.0f - iou + v + 1e-10f);
  float ciou  = iou - (cd / diag + alpha * v);
  float box_sum = 1.0f - ciou;

  // WMMA wave reductions (uniform flow -> EXEC all ones), then LDS across waves.
  float cls_w = wave_sum_wmma(cls_sum);
  float box_w = wave_sum_wmma(box_sum);
  int wid = threadIdx.x >> 5;
  if ((threadIdx.x & 31) == 0) { smem[wid] = cls_w; smem[8 + wid] = box_w; }
  __syncthreads();
  if (threadIdx.x == 0) {
    float cs = 0.0f, bs = 0.0f;
    for (int w = 0; w < 8; ++w) { cs += smem[w]; bs += smem[8 + w]; }
    partials[300 + blockIdx.x] = cs;   // cls partial
    partials[302 + blockIdx.x] = bs;   // box partial
  }
}

// Objectness BCE over all B*A*H*W cells, vectorized: each thread owns one
// 16B-aligned float4 of an objectness plane (planes are 6400 contiguous
// floats, 6400 % 4 == 0 -> alignment holds) plus one mask word (a 4-aligned
// bit quartet never straddles a 32-bit word). 300 blocks x 256 threads =
// 76800 threads = exactly NVEC chunks; uniform flow keeps EXEC all-ones
// for the WMMA reduction.
__global__ void obj_kernel(const float* __restrict__ pred,
                           const unsigned int* __restrict__ mask,
                           float* __restrict__ partials) {
  __shared__ float smem[8];
  int vid = blockIdx.x * blockDim.x + threadIdx.x;   // 0 .. NVEC-1
  int ba  = vid / (PLANEc / 4);                      // which (b,a) plane
  int r   = vid - ba * (PLANEc / 4);                 // float4 index in plane

  const v4f* p4 = (const v4f*)(pred + ((long)ba * CHc + 4) * PLANEc);
  v4f x = p4[r];                                     // GLOBAL_LOAD_B128

  int cell0 = ba * PLANEc + r * 4;                   // first of 4 cells
  unsigned int mw  = mask[cell0 >> 5];
  unsigned int sh  = (unsigned int)(cell0 & 31);

  float s = 0.0f;
  s += bce_logits(x[0], (float)((mw >> (sh + 0)) & 1u));
  s += bce_logits(x[1], (float)((mw >> (sh + 1)) & 1u));
  s += bce_logits(x[2], (float)((mw >> (sh + 2)) & 1u));
  s += bce_logits(x[3], (float)((mw >> (sh + 3)) & 1u));

  float ws2 = wave_sum_wmma(s);
  int wid = threadIdx.x >> 5;
  if ((threadIdx.x & 31) == 0) smem[wid] = ws2;
  __syncthreads();
  if (threadIdx.x == 0) {
    float bs = 0.0f;
    for (int w = 0; w < 8; ++w) bs += smem[w];
    partials[blockIdx.x] = bs;
  }
}

// Fixed-order final sum: bitwise-deterministic across graph replays.
__global__ void finalize_kernel(const float* __restrict__ partials,
                                float* __restrict__ out) {
  float obj_s = 0.0f;
  for (int i = 0; i < OBJ_BLOCKS; ++i) obj_s += partials[i];
  float cls_s = partials[300] + partials[301];
  float box_s = partials[302] + partials[303];
  float box_loss = box_s / (float)Nc;
  float cls_loss = cls_s / (float)Nc;
  float obj_loss = obj_s / (float)NCELL;
  out[0] = 5.0f * box_loss + obj_loss + cls_loss;
  out[1] = box_loss;
  out[2] = obj_loss;
  out[3] = cls_loss;
}

extern "C" void kernel_launch(void* const* d_in, const int* in_sizes, int n_in,
                              void* d_out, int out_size, void* d_ws, size_t ws_size,
                              hipStream_t stream) {
  const float* pred    = (const float*)d_in[0];   // (16, 255, 80, 80) f32
  const float* targets = (const float*)d_in[1];   // (512, 6) f32
  const float* anchors = (const float*)d_in[2];   // (3, 2) f32

  float*        partials = (float*)d_ws;
  unsigned int* mask     = (unsigned int*)((char*)d_ws + WS_MASK_BYTE_OFF);

  zero_ws_kernel<<<40, 256, 0, stream>>>(mask);
  per_target_kernel<<<Nc / 256, 256, 0, stream>>>(pred, targets, anchors, mask, partials);
  obj_kernel<<<OBJ_BLOCKS, 256, 0, stream>>>(pred, mask, partials);
  finalize_kernel<<<1, 1, 0, stream>>>(partials, (float*)d_out);
}